// RasterizationModel_6141803233877
// MI455X (gfx1250) — compile-verified
//
#include <hip/hip_runtime.h>
#include <math.h>

#define NG    1536
#define IMG_W 160
#define IMG_H 128
#define FXV   146.44f
#define FYV   146.44f
#define CXV   80.0f
#define CYV   64.0f
#define EPS2D 0.3f
#define NEARV 0.01f
#define FARV  1.0e10f
#define NPAD  2048
#define NTILE (NG / 16)          // 96 gaussian tiles
#define NSTRIP (IMG_H * (IMG_W / 16))  // 1280 pixel strips

typedef __attribute__((ext_vector_type(2))) float v2f;
typedef __attribute__((ext_vector_type(8))) float v8f;

// ---------------------------------------------------------------------------
// Kernel 1: per-gaussian preprocessing -> quadratic-form features + colors
// ---------------------------------------------------------------------------
__global__ __launch_bounds__(256) void preprocess_kernel(
    const float* __restrict__ means, const float* __restrict__ quats,
    const float* __restrict__ log_scales, const float* __restrict__ opacity_logits,
    const float* __restrict__ sh, const float* __restrict__ c2w,
    float* __restrict__ depthkey, float* __restrict__ qfeat,
    float* __restrict__ colors)
{
  int n = blockIdx.x * blockDim.x + threadIdx.x;
  if (n >= NG) return;

  float cw[12];
#pragma unroll
  for (int i = 0; i < 12; ++i) cw[i] = c2w[i];

  // R_w2c[i][j] = c2w[j][i] * (i==0 ? 1 : -1)   (columns 1,2 flipped, then T)
  float R[3][3];
#pragma unroll
  for (int i = 0; i < 3; ++i)
#pragma unroll
    for (int j = 0; j < 3; ++j)
      R[i][j] = cw[j * 4 + i] * (i == 0 ? 1.0f : -1.0f);
  float cpx = cw[3], cpy = cw[7], cpz = cw[11];
  float tw[3];
#pragma unroll
  for (int i = 0; i < 3; ++i)
    tw[i] = -(R[i][0] * cpx + R[i][1] * cpy + R[i][2] * cpz);

  float mx = means[n * 3 + 0], my = means[n * 3 + 1], mz = means[n * 3 + 2];
  float pcx = R[0][0] * mx + R[0][1] * my + R[0][2] * mz + tw[0];
  float pcy = R[1][0] * mx + R[1][1] * my + R[1][2] * mz + tw[1];
  float pcz = R[2][0] * mx + R[2][1] * my + R[2][2] * mz + tw[2];
  float depth = pcz;
  bool mask = (depth > NEARV) && (depth < FARV);

  // view direction + SH basis (16 terms)
  float dxr = mx - cpx, dyr = my - cpy, dzr = mz - cpz;
  float invn = rsqrtf(dxr * dxr + dyr * dyr + dzr * dzr);
  float x = dxr * invn, y = dyr * invn, z = dzr * invn;
  float b[16];
  b[0]  = 0.282095f;
  b[1]  = -0.488603f * y;
  b[2]  = 0.488603f * z;
  b[3]  = -0.488603f * x;
  b[4]  = 1.092548f * x * y;
  b[5]  = -1.092548f * y * z;
  b[6]  = 0.315392f * (3.0f * z * z - 1.0f);
  b[7]  = -1.092548f * x * z;
  b[8]  = 0.546274f * (x * x - y * y);
  b[9]  = -0.590044f * y * (3.0f * x * x - y * y);
  b[10] = 2.890611f * x * y * z;
  b[11] = -0.457046f * y * (5.0f * z * z - 1.0f);
  b[12] = 0.373176f * z * (5.0f * z * z - 3.0f);
  b[13] = -0.457046f * x * (5.0f * z * z - 1.0f);
  b[14] = 1.445306f * z * (x * x - y * y);
  b[15] = -0.590044f * x * (x * x - 3.0f * y * y);
  float col[3];
#pragma unroll
  for (int c = 0; c < 3; ++c) {
    float s = 0.0f;
#pragma unroll
    for (int k = 0; k < 16; ++k) s += b[k] * sh[n * 48 + k * 3 + c];
    col[c] = fmaxf(s + 0.5f, 0.0f);
  }

  // quat -> rotation, scaled columns, 3D covariance
  float qw = quats[n * 4 + 0], qx = quats[n * 4 + 1];
  float qy = quats[n * 4 + 2], qz = quats[n * 4 + 3];
  float qn = rsqrtf(qw * qw + qx * qx + qy * qy + qz * qz);
  qw *= qn; qx *= qn; qy *= qn; qz *= qn;
  float Rg[3][3] = {
    {1.0f - 2.0f * (qy * qy + qz * qz), 2.0f * (qx * qy - qz * qw), 2.0f * (qx * qz + qy * qw)},
    {2.0f * (qx * qy + qz * qw), 1.0f - 2.0f * (qx * qx + qz * qz), 2.0f * (qy * qz - qx * qw)},
    {2.0f * (qx * qz - qy * qw), 2.0f * (qy * qz + qx * qw), 1.0f - 2.0f * (qx * qx + qy * qy)}};
  float s0 = __expf(log_scales[n * 3 + 0]);
  float s1 = __expf(log_scales[n * 3 + 1]);
  float s2 = __expf(log_scales[n * 3 + 2]);
  float M[3][3];
#pragma unroll
  for (int i = 0; i < 3; ++i) {
    M[i][0] = Rg[i][0] * s0; M[i][1] = Rg[i][1] * s1; M[i][2] = Rg[i][2] * s2;
  }
  float C3[3][3];
#pragma unroll
  for (int i = 0; i < 3; ++i)
#pragma unroll
    for (int j = 0; j < 3; ++j)
      C3[i][j] = M[i][0] * M[j][0] + M[i][1] * M[j][1] + M[i][2] * M[j][2];
  // cov_cam = R * C3 * R^T
  float RC[3][3];
#pragma unroll
  for (int i = 0; i < 3; ++i)
#pragma unroll
    for (int j = 0; j < 3; ++j)
      RC[i][j] = R[i][0] * C3[0][j] + R[i][1] * C3[1][j] + R[i][2] * C3[2][j];
  float CC[3][3];
#pragma unroll
  for (int i = 0; i < 3; ++i)
#pragma unroll
    for (int j = 0; j < 3; ++j)
      CC[i][j] = RC[i][0] * R[j][0] + RC[i][1] * R[j][1] + RC[i][2] * R[j][2];

  float rz = 1.0f / pcz;
  float J00 = FXV * rz, J02 = -FXV * pcx * rz * rz;
  float J11 = FYV * rz, J12 = -FYV * pcy * rz * rz;
  float JC0[3], JC1[3];
#pragma unroll
  for (int k = 0; k < 3; ++k) {
    JC0[k] = J00 * CC[0][k] + J02 * CC[2][k];
    JC1[k] = J11 * CC[1][k] + J12 * CC[2][k];
  }
  float c2_00 = JC0[0] * J00 + JC0[2] * J02;
  float c2_01 = JC0[1] * J11 + JC0[2] * J12;
  float c2_11 = JC1[1] * J11 + JC1[2] * J12;
  float a  = c2_00 + EPS2D;
  float cc = c2_11 + EPS2D;
  float bb = c2_01;
  float idet = 1.0f / (a * cc - bb * bb);
  float Aq = cc * idet, Bq = -bb * idet, Cq = a * idet;
  float u = FXV * pcx * rz + CXV;
  float v = FYV * pcy * rz + CYV;
  float opac = 1.0f / (1.0f + __expf(-opacity_logits[n]));

  // power(n, pixel) = q . [x^2, y^2, xy, x, y, 1, 0, 0]; opacity folded via log
  float q[8];
  q[0] = -0.5f * Aq;
  q[1] = -0.5f * Cq;
  q[2] = -Bq;
  q[3] = Aq * u + Bq * v;
  q[4] = Cq * v + Bq * u;
  q[5] = -0.5f * (Aq * u * u + Cq * v * v) - Bq * u * v + __logf(opac);
  q[6] = 0.0f;
  q[7] = 0.0f;
  if (!mask) { q[0] = q[1] = q[2] = q[3] = q[4] = 0.0f; q[5] = -1e30f; }
#pragma unroll
  for (int k = 0; k < 8; ++k) qfeat[n * 8 + k] = q[k];
#pragma unroll
  for (int c = 0; c < 3; ++c) colors[n * 3 + c] = col[c];
  depthkey[n] = mask ? depth : __builtin_inff();
}

// ---------------------------------------------------------------------------
// Kernel 2: single-block bitonic argsort by depth + gather into sorted arrays
// ---------------------------------------------------------------------------
__global__ __launch_bounds__(1024) void sort_gather_kernel(
    const float* __restrict__ key_in, const float* __restrict__ qfeat,
    const float* __restrict__ colors, float* __restrict__ sq,
    float* __restrict__ sc)
{
  __shared__ float key[NPAD];
  __shared__ int   idx[NPAD];
  int tid = threadIdx.x;
  for (int i = tid; i < NPAD; i += 1024) {
    key[i] = (i < NG) ? key_in[i] : __builtin_inff();
    idx[i] = i;
  }
  for (int k = 2; k <= NPAD; k <<= 1) {
    for (int j = k >> 1; j > 0; j >>= 1) {
      __syncthreads();
      for (int i = tid; i < NPAD; i += 1024) {
        int ixj = i ^ j;
        if (ixj > i) {
          bool up = ((i & k) == 0);
          float ki = key[i], kj = key[ixj];
          bool swap = up ? (ki > kj) : (ki < kj);
          if (swap) {
            key[i] = kj; key[ixj] = ki;
            int t = idx[i]; idx[i] = idx[ixj]; idx[ixj] = t;
          }
        }
      }
    }
  }
  __syncthreads();
  for (int i = tid; i < NG; i += 1024) {
    int g = idx[i];
    if (g < NG) {
#pragma unroll
      for (int k = 0; k < 8; ++k) sq[i * 8 + k] = qfeat[g * 8 + k];
#pragma unroll
      for (int c = 0; c < 3; ++c) sc[i * 3 + c] = colors[g * 3 + c];
    } else {  // padding slot leaked into first NG positions: make it invisible
#pragma unroll
      for (int k = 0; k < 8; ++k) sq[i * 8 + k] = 0.0f;
      sq[i * 8 + 5] = -1e30f;
#pragma unroll
      for (int c = 0; c < 3; ++c) sc[i * 3 + c] = 0.0f;
    }
  }
}

// ---------------------------------------------------------------------------
// Kernel 3: WMMA rasterizer. One wave32 per 16-pixel strip.
//   power tile (16 gaussians x 16 pixels) = two chained v_wmma_f32_16x16x4_f32
// ---------------------------------------------------------------------------
__global__ __launch_bounds__(128) void raster_kernel(
    const float* __restrict__ sq, const float* __restrict__ sc,
    float* __restrict__ out)
{
  int lane = threadIdx.x & 31;
  int wave = threadIdx.x >> 5;
  int strip = blockIdx.x * 4 + wave;            // 0 .. NSTRIP-1
  int y  = strip / (IMG_W / 16);
  int x0 = (strip % (IMG_W / 16)) * 16;
  int hi = lane >> 4;                           // 0: gaussians 0-7, 1: 8-15
  int pl = lane & 15;                           // pixel within strip
  float px = (float)(x0 + pl);
  float py = (float)y;

  // B operand: pixel monomials f = [x^2, y^2, xy, x | y, 1, 0, 0]
  // 4x16 layout: lanes 0-15 hold {K0,K1}, lanes 16-31 hold {K2,K3} per wmma.
  v2f b0, b1;
  if (!hi) { b0.x = px * px; b0.y = py * py; b1.x = py;   b1.y = 1.0f; }
  else     { b0.x = px * py; b0.y = px;      b1.x = 0.0f; b1.y = 0.0f; }

  float T = 1.0f;
  float cr = 0.0f, cg = 0.0f, cb = 0.0f;

  for (int tile = 0; tile < NTILE; ++tile) {
    // A operand: 16x4 layout -> lane m holds {K0,K1}, lane m+16 holds {K2,K3}
    int gl = tile * 16 + pl;
    const float* qb = sq + gl * 8 + (hi ? 2 : 0);
    v2f a0 = *(const v2f*)(qb);      // K0,K1 (lo) / K2,K3 (hi)
    v2f a1 = *(const v2f*)(qb + 4);  // K4,K5 (lo) / K6,K7 (hi)

    v8f d = {};
    d = __builtin_amdgcn_wmma_f32_16x16x4_f32(false, a0, false, b0,
                                              (short)0, d, false, false);
    d = __builtin_amdgcn_wmma_f32_16x16x4_f32(false, a1, false, b1,
                                              (short)0, d, false, false);
    // D layout: VGPR r, lanes 0-15 -> gaussian r; lanes 16-31 -> gaussian r+8
    int gbase = tile * 16 + hi * 8;
    float t_part = 1.0f;
    float pr = 0.0f, pg = 0.0f, pb = 0.0f;
#pragma unroll
    for (int r = 0; r < 8; ++r) {
      float alpha = fminf(0.999f, __expf(d[r]));
      const float* col = sc + (gbase + r) * 3;
      float w = t_part * alpha;
      pr += w * col[0];
      pg += w * col[1];
      pb += w * col[2];
      t_part *= (1.0f - alpha);
    }
    // merge lo/hi halves (sorted order: lo gaussians precede hi)
    float t_h  = __shfl(t_part, lane ^ 16, 32);
    float pr_h = __shfl(pr,     lane ^ 16, 32);
    float pg_h = __shfl(pg,     lane ^ 16, 32);
    float pb_h = __shfl(pb,     lane ^ 16, 32);
    if (!hi) {
      cr += T * (pr + t_part * pr_h);
      cg += T * (pg + t_part * pg_h);
      cb += T * (pb + t_part * pb_h);
      T  *= t_part * t_h;
    }
    // deterministic early-out once every pixel in the strip is saturated
    bool active = (!hi) && (T > 1e-6f);
    if (!__any((int)active)) break;
  }

  if (!hi) {
    int o = (y * IMG_W + x0 + pl) * 3;
    out[o + 0] = cr;
    out[o + 1] = cg;
    out[o + 2] = cb;
  }
}

// ---------------------------------------------------------------------------
extern "C" void kernel_launch(void* const* d_in, const int* in_sizes, int n_in,
                              void* d_out, int out_size, void* d_ws, size_t ws_size,
                              hipStream_t stream) {
  const float* means          = (const float*)d_in[0];
  const float* quats          = (const float*)d_in[1];
  const float* log_scales     = (const float*)d_in[2];
  const float* opacity_logits = (const float*)d_in[3];
  const float* sh             = (const float*)d_in[4];
  const float* c2w            = (const float*)d_in[5];
  float* out = (float*)d_out;

  float* ws       = (float*)d_ws;
  float* depthkey = ws;                 // NG
  float* qfeat    = depthkey + NG;      // NG*8
  float* colors   = qfeat + NG * 8;     // NG*3
  float* sq       = colors + NG * 3;    // NG*8 (sorted)
  float* sc       = sq + NG * 8;        // NG*3 (sorted)

  preprocess_kernel<<<(NG + 255) / 256, 256, 0, stream>>>(
      means, quats, log_scales, opacity_logits, sh, c2w,
      depthkey, qfeat, colors);
  sort_gather_kernel<<<1, 1024, 0, stream>>>(depthkey, qfeat, colors, sq, sc);
  raster_kernel<<<NSTRIP / 4, 128, 0, stream>>>(sq, sc, out);
}